// HOGLayer_43344809951565
// MI455X (gfx1250) — compile-verified
//
#include <hip/hip_runtime.h>
#include <math.h>

// ---------------------------------------------------------------------------
// HOG layer fused kernel for MI455X (gfx1250, wave32).
//   out[n,c,i,j] = (1/64) * sum_{8x8 window, pad 1} bins[n,c,·,·]
//   bins from Sobel magnitude/phase soft-binning (10 bins over pi).
// Separable 8x8 box filter as two banded-ones matmuls on the matrix pipe via
// V_WMMA_F32_16X16X4_F32 (full f32, like the reference). Input patch staged
// to LDS with GLOBAL_LOAD_ASYNC_TO_LDS_B32 on interior tiles (ASYNCcnt).
// All phase-3 LDS reads are unguarded via zero-padded bins patch -> no
// exec-mask save/restore in the WMMA inner loops.
// ---------------------------------------------------------------------------

typedef float v2f __attribute__((ext_vector_type(2)));
typedef float v8f __attribute__((ext_vector_type(8)));

#define HW      512
#define OHW     507          // (512 + 2*1 - 8) + 1
#define NB      10
#define TILE    16
#define BINS_D  23           // valid extent: 16 outputs + 7 window tail
#define BINS_R  24           // allocated rows (K padded to 6*4)
#define BINS_S  32           // allocated col stride (covers B col group 1)
#define INP_D   25           // bins + sobel halo
#define INP_S   28           // padded row stride
#define TS_S    32
#define WAVE_LDS_F (INP_D * INP_S + NB * BINS_R * BINS_S + TILE * TS_S) // 8892

__device__ __forceinline__ float mod10f(float x) {
    // jnp.mod semantics (sign of divisor; divisor = +10)
    return x - floorf(x * 0.1f) * 10.0f;
}

// gfx1250 async copy: one f32 element global -> LDS, tracked by ASYNCcnt.
__device__ __forceinline__ void async_f32_to_lds(unsigned lds_byte_off,
                                                 const float* gptr) {
    asm volatile("global_load_async_to_lds_b32 %0, %1, off"
                 :
                 : "v"(lds_byte_off),
                   "v"((unsigned long long)(__SIZE_TYPE__)gptr)
                 : "memory");
}
__device__ __forceinline__ void wait_asynccnt0() {
    asm volatile("s_wait_asynccnt 0" ::: "memory");
}

__global__ __launch_bounds__(32)
void hog_fused_kernel(const float* __restrict__ x, float* __restrict__ out) {
    __shared__ float lds[WAVE_LDS_F];

    const unsigned lane = threadIdx.x & 31u;
    const int img  = blockIdx.y;                // 0..31
    const int tile = blockIdx.x;                // 0..1023
    const int ty   = tile >> 5;
    const int tx   = tile & 31;
    const int i0   = ty * TILE;                 // first output row
    const int j0   = tx * TILE;                 // first output col

    float* inT   = lds;                         // [25][28]
    float* binsT = inT + INP_D * INP_S;         // [10][24][32] zero-padded
    float* Ts    = binsT + NB * BINS_R * BINS_S;// [16][32]

    const float* xim = x + (size_t)img * HW * HW;

    // ------------------------------------------------------------------
    // Phase 0: zero the padded bins patch (float4 stores).
    // ------------------------------------------------------------------
    {
        float4 z4 = make_float4(0.f, 0.f, 0.f, 0.f);
        float4* bz = (float4*)binsT;
        const int n4 = NB * BINS_R * BINS_S / 4;        // 1920
        for (int e = lane; e < n4; e += 32) bz[e] = z4;
    }

    // ------------------------------------------------------------------
    // Phase 1: stage 25x25 input patch (zero padded) into LDS.
    // Interior tiles: async LDS loads (wave-uniform branch, 1 wave/block).
    // ------------------------------------------------------------------
    const int r0 = i0 - 2, c0 = j0 - 2;
    const bool interior = (r0 >= 0) && (c0 >= 0) &&
                          (r0 + INP_D <= HW) && (c0 + INP_D <= HW);
    if (interior) {
        __builtin_prefetch(xim + (size_t)r0 * HW + c0, 0, 1); // global_prefetch_b8
        const unsigned base = (unsigned)(__SIZE_TYPE__)inT;
        for (int e = lane; e < INP_D * INP_D; e += 32) {
            int r = e / INP_D, c = e % INP_D;
            async_f32_to_lds(base + (unsigned)(r * INP_S + c) * 4u,
                             xim + (size_t)(r0 + r) * HW + (c0 + c));
        }
        wait_asynccnt0();
    } else {
        for (int e = lane; e < INP_D * INP_D; e += 32) {
            int r = e / INP_D, c = e % INP_D;
            int ar = r0 + r, ac = c0 + c;
            float v = 0.0f;
            if (ar >= 0 && ar < HW && ac >= 0 && ac < HW)
                v = xim[(size_t)ar * HW + ac];
            inT[r * INP_S + c] = v;
        }
    }
    __syncthreads();

    // ------------------------------------------------------------------
    // Phase 2: Sobel + atan2 + soft binning -> 10-channel bins patch.
    // bins local (r,c) maps to absolute (i0-1+r, j0-1+c); OOB pixels = 0
    // (already zeroed in phase 0, but stored anyway for uniformity).
    // ------------------------------------------------------------------
    for (int e = lane; e < BINS_D * BINS_D; e += 32) {
        int r = e / BINS_D, c = e % BINS_D;
        int ar = i0 - 1 + r, ac = j0 - 1 + c;
        float vb[NB];
        #pragma unroll
        for (int k = 0; k < NB; ++k) vb[k] = 0.0f;
        if (ar >= 0 && ar < HW && ac >= 0 && ac < HW) {
            const float* p = inT + (r + 1) * INP_S + (c + 1); // center
            float a00 = p[-INP_S - 1], a01 = p[-INP_S], a02 = p[-INP_S + 1];
            float a10 = p[-1],                          a12 = p[1];
            float a20 = p[ INP_S - 1], a21 = p[ INP_S], a22 = p[ INP_S + 1];
            // weight[0] = [[1,0,-1],[2,0,-2],[1,0,-1]] ; weight[1] = its T
            float g0 = (a00 - a02) + 2.0f * (a10 - a12) + (a20 - a22);
            float g1 = (a00 + 2.0f * a01 + a02) - (a20 + 2.0f * a21 + a22);
            float nrm = sqrtf(g0 * g0 + g1 * g1);
            float ph  = atan2f(g0, g1);                 // reference arg order
            float pi_ = ph * ((float)NB / (float)M_PI);
            float fb = floorf(pi_), ft = ceilf(pi_);
            float b = mod10f(fb), t = mod10f(ft), f = mod10f(pi_);
            float t_v = nrm * (1.0f - (t - f));
            float b_v = nrm * (1.0f - (f - b));
            int ib = (int)fb % NB; if (ib < 0) ib += NB;
            int it = (int)ft % NB; if (it < 0) it += NB;
            vb[ib] += b_v;
            vb[it] += t_v;                              // ib==it -> both add
        }
        #pragma unroll
        for (int k = 0; k < NB; ++k)
            binsT[(k * BINS_R + r) * BINS_S + c] = vb[k];
    }
    __syncthreads();

    // ------------------------------------------------------------------
    // Phase 3: separable 8-wide box sums on the matrix pipe.
    //   T  = P (16x24 banded ones) x bins_c (24x32 zero-padded)
    //   Out = T (16x24) x Q (24x16 banded ones)
    // All LDS reads in-bounds by construction (padding is zero), so no
    // per-lane exec masking around ds loads.
    // WMMA f32 16x16x4 lane layout:
    //   A 16x4 : M = lane&15 ; VGPR{0,1} = K {0,1} (lanes<16) / {2,3} (>=16)
    //   B 4x16 : N = lane&15 ; VGPR{0,1} = K {0,1} (lanes<16) / {2,3} (>=16)
    //   C/D    : N = lane&15 ; VGPR v    = M  v    (lanes<16) /  v+8 (>=16)
    // ------------------------------------------------------------------
    const int mA  = lane & 15;
    const int kHi = (lane < 16) ? 0 : 2;
    const int mD  = (lane < 16) ? 0 : 8;

    const float invK = 1.0f / 64.0f;
    float* oim = out + (size_t)img * NB * OHW * OHW;

    // Band operands are channel-invariant; computed once, reused in-loop.
    v2f Aband[6], Bband[6];
    #pragma unroll
    for (int kt = 0; kt < 6; ++kt) {
        int k0 = kt * 4 + kHi;
        Aband[kt].x = (k0     >= mA && k0     <= mA + 7) ? 1.0f : 0.0f;
        Aband[kt].y = (k0 + 1 >= mA && k0 + 1 <= mA + 7) ? 1.0f : 0.0f;
        Bband[kt] = Aband[kt];  // same band, role of (row,col) mirrored
    }

    for (int ch = 0; ch < NB; ++ch) {
        const float* bc = binsT + ch * BINS_R * BINS_S;
        __syncthreads(); // protect Ts reuse across channel iterations

        // ---- vertical pass: two 16-column groups ----
        #pragma unroll
        for (int g = 0; g < 2; ++g) {
            v8f acc = {};
            const int col = g * 16 + mA;                 // 0..31, in-bounds
            #pragma unroll
            for (int kt = 0; kt < 6; ++kt) {
                int k0 = kt * 4 + kHi;
                v2f B;
                B.x = bc[ k0      * BINS_S + col];
                B.y = bc[(k0 + 1) * BINS_S + col];
                acc = __builtin_amdgcn_wmma_f32_16x16x4_f32(
                    false, Aband[kt], false, B, (short)0, acc, false, false);
            }
            #pragma unroll
            for (int v = 0; v < 8; ++v)
                Ts[(mD + v) * TS_S + col] = acc[v];      // unconditional
        }
        __syncthreads(); // Ts D-layout -> A-layout round trip

        // ---- horizontal pass ----
        v8f acc2 = {};
        #pragma unroll
        for (int kt = 0; kt < 6; ++kt) {
            int k0 = kt * 4 + kHi;
            v2f A;
            A.x = Ts[mA * TS_S + k0];                    // col 23 == 0 exact
            A.y = Ts[mA * TS_S + k0 + 1];
            acc2 = __builtin_amdgcn_wmma_f32_16x16x4_f32(
                false, A, false, Bband[kt], (short)0, acc2, false, false);
        }
        #pragma unroll
        for (int v = 0; v < 8; ++v) {
            int row = i0 + mD + v;
            int colO = j0 + mA;
            if (row < OHW && colO < OHW)
                oim[((size_t)ch * OHW + row) * OHW + colO] = acc2[v] * invK;
        }
    }
}

extern "C" void kernel_launch(void* const* d_in, const int* in_sizes, int n_in,
                              void* d_out, int out_size, void* d_ws, size_t ws_size,
                              hipStream_t stream) {
    (void)in_sizes; (void)n_in; (void)d_ws; (void)ws_size; (void)out_size;
    const float* x = (const float*)d_in[0];
    // d_in[1] is the fixed Sobel stack; its values are baked into the kernel.
    float* out = (float*)d_out;

    // 1024 tiles (32x32 of 16x16 outputs) per image, 1 wave (tile) per block.
    dim3 grid(1024, 32);
    dim3 block(32);
    hog_fused_kernel<<<grid, block, 0, stream>>>(x, out);
}